// SphericalHarmonics_57440892617149
// MI455X (gfx1250) — compile-verified
//
#include <hip/hip_runtime.h>

typedef __attribute__((ext_vector_type(2))) float v2f;
typedef __attribute__((ext_vector_type(8))) float v8f;

#define ISQ4PI 0.28209479177387814f   // 1/sqrt(4*pi)
#define S3     1.7320508075688772f    // sqrt(3)
#define S5     2.2360679774997896f    // sqrt(5)
#define S7     2.6457513110645907f    // sqrt(7)
#define S15    3.8729833462074170f    // sqrt(15)
#define S105   10.246950765959598f    // sqrt(105)
#define SA     4.1833001326703830f    // sqrt(630)/6
#define S168_8 1.6201851746019653f    // sqrt(168)/8

// Coefficient matrix C[20][16]: monomial basis (K order)
//   k: 0:1  1:x  2:y  3:z  4:x2 5:y2 6:z2 7:xy 8:xz 9:yz
//      10:x3 11:x2y 12:x2z 13:xy2 14:xyz 15:xz2 16:y3 17:y2z 18:yz2 19:z3
// -> 16 SH components (e3nn order), 1/sqrt(4pi) folded in.
struct CMat { float v[20][16]; };

constexpr CMat make_cm() {
  CMat m{};
  m.v[0][0]   = 1.0f * ISQ4PI;               // sh0
  m.v[1][1]   = S3 * ISQ4PI;                 // sh1_0 = s3*x
  m.v[2][2]   = S3 * ISQ4PI;                 // sh1_1 = s3*y
  m.v[3][3]   = S3 * ISQ4PI;                 // sh1_2 = s3*z
  m.v[8][4]   = S15 * ISQ4PI;                // sh2_0 = s15*xz
  m.v[7][5]   = S15 * ISQ4PI;                // sh2_1 = s15*xy
  m.v[5][6]   = S5 * ISQ4PI;                 // sh2_2 = s5*(y2 - .5x2 - .5z2)
  m.v[4][6]   = -0.5f * S5 * ISQ4PI;
  m.v[6][6]   = -0.5f * S5 * ISQ4PI;
  m.v[9][7]   = S15 * ISQ4PI;                // sh2_3 = s15*yz
  m.v[6][8]   = 0.5f * S15 * ISQ4PI;         // sh2_4 = .5*s15*(z2-x2)
  m.v[4][8]   = -0.5f * S15 * ISQ4PI;
  m.v[15][9]  = 1.5f * SA * ISQ4PI;          // sh3_0
  m.v[10][9]  = -0.5f * SA * ISQ4PI;
  m.v[14][10] = S105 * ISQ4PI;               // sh3_1
  m.v[13][11] = 4.0f * S168_8 * ISQ4PI;      // sh3_2
  m.v[10][11] = -S168_8 * ISQ4PI;
  m.v[15][11] = -S168_8 * ISQ4PI;
  m.v[16][12] = S7 * ISQ4PI;                 // sh3_3
  m.v[11][12] = -1.5f * S7 * ISQ4PI;
  m.v[18][12] = -1.5f * S7 * ISQ4PI;
  m.v[17][13] = 4.0f * S168_8 * ISQ4PI;      // sh3_4
  m.v[12][13] = -S168_8 * ISQ4PI;
  m.v[19][13] = -S168_8 * ISQ4PI;
  m.v[18][14] = 0.5f * S105 * ISQ4PI;        // sh3_5
  m.v[11][14] = -0.5f * S105 * ISQ4PI;
  m.v[19][15] = 0.5f * SA * ISQ4PI;          // sh3_6
  m.v[12][15] = -1.5f * SA * ISQ4PI;
  return m;
}

__constant__ CMat CM = make_cm();

__global__ __launch_bounds__(256) void SphericalHarmonics_57440892617149_kernel(
    const float* __restrict__ x, float* __restrict__ out, int n) {
  const int lane   = (int)(threadIdx.x & 31u);
  const int col    = lane & 15;          // SH component / point-in-tile for loads
  const bool hiH   = lane >= 16;         // half-wave: K slice + row block
  const int hi2    = hiH ? 2 : 0;

  const unsigned wavesPerBlock = blockDim.x >> 5;
  const unsigned waveId = blockIdx.x * wavesPerBlock + (threadIdx.x >> 5);
  const unsigned nWaves = gridDim.x * wavesPerBlock;
  const unsigned nTiles = ((unsigned)n + 15u) >> 4;

  // Loop-invariant B operands (coefficient matrix), per-lane.
  // B 4x16 f32 layout mirrors A: vgpr0 = row (4j + 0|2), vgpr1 = row (4j + 1|3).
  float b0[5], b1[5];
#pragma unroll
  for (int j = 0; j < 5; ++j) {
    b0[j] = CM.v[4 * j + hi2][col];
    b1[j] = CM.v[4 * j + 1 + hi2][col];
  }

  for (unsigned t = waveId; t < nTiles; t += nWaves) {
    const unsigned p0 = t << 4;

    // Prefetch next tile's input (wave-uniform trip; global_prefetch_b8).
    if (t + nWaves < nTiles) {
      __builtin_prefetch(x + 3u * (((t + nWaves) << 4) + (unsigned)col), 0, 0);
    }

    // Both half-waves load the same 16 points (A duplication across K slices).
    unsigned p = p0 + (unsigned)col;
    if (p >= (unsigned)n) p = (unsigned)n - 1u;   // clamp: EXEC stays all-1s
    const unsigned xo = 3u * p;                   // 32-bit offset -> SADDR form
    float vx = __builtin_nontemporal_load(x + xo);
    float vy = __builtin_nontemporal_load(x + xo + 1u);
    float vz = __builtin_nontemporal_load(x + xo + 2u);
#if __has_builtin(__builtin_amdgcn_rsqf)
    const float rinv = __builtin_amdgcn_rsqf(vx * vx + vy * vy + vz * vz);
#else
    const float rinv = rsqrtf(vx * vx + vy * vy + vz * vz);
#endif
    vx *= rinv; vy *= rinv; vz *= rinv;

    // Monomials (named scalars -> clean SROA, no indexed-extract trees).
    const float xx = vx * vx, yy = vy * vy, zz = vz * vz;
    const float xy = vx * vy, xz = vx * vz, yz = vy * vz;
    const float x3 = xx * vx, x2y = xx * vy, x2z = xx * vz;
    const float xy2 = yy * vx, xyz = xy * vz, xz2 = zz * vx;
    const float y3 = yy * vy, y2z = yy * vz, yz2 = zz * vy, z3 = zz * vz;

    // Per-chunk A operands: low half supplies K=4j,4j+1; high half K=4j+2,4j+3.
    const float a00 = hiH ? vy  : 1.0f, a01 = hiH ? vz  : vx;   // {1,x | y,z}
    const float a10 = hiH ? zz  : xx,   a11 = hiH ? xy  : yy;   // {x2,y2 | z2,xy}
    const float a20 = hiH ? x3  : xz,   a21 = hiH ? x2y : yz;   // {xz,yz | x3,x2y}
    const float a30 = hiH ? xyz : x2z,  a31 = hiH ? xz2 : xy2;  // {x2z,xy2 | xyz,xz2}
    const float a40 = hiH ? yz2 : y3,   a41 = hiH ? z3  : y2z;  // {y3,y2z | yz2,z3}

    // D[16 pts x 16 comps] = sum_j A_j[16x4] * B_j[4x16], K=20 over 5 WMMAs.
    v8f acc = {0.f, 0.f, 0.f, 0.f, 0.f, 0.f, 0.f, 0.f};
    v2f a, b;
    a.x = a00; a.y = a01; b.x = b0[0]; b.y = b1[0];
    acc = __builtin_amdgcn_wmma_f32_16x16x4_f32(false, a, false, b, (short)0, acc, false, false);
    a.x = a10; a.y = a11; b.x = b0[1]; b.y = b1[1];
    acc = __builtin_amdgcn_wmma_f32_16x16x4_f32(false, a, false, b, (short)0, acc, false, false);
    a.x = a20; a.y = a21; b.x = b0[2]; b.y = b1[2];
    acc = __builtin_amdgcn_wmma_f32_16x16x4_f32(false, a, false, b, (short)0, acc, false, false);
    a.x = a30; a.y = a31; b.x = b0[3]; b.y = b1[3];
    acc = __builtin_amdgcn_wmma_f32_16x16x4_f32(false, a, false, b, (short)0, acc, false, false);
    a.x = a40; a.y = a41; b.x = b0[4]; b.y = b1[4];
    acc = __builtin_amdgcn_wmma_f32_16x16x4_f32(false, a, false, b, (short)0, acc, false, false);

    // D layout: lane holds component `col` of rows (hi*8 + r), r = vgpr index.
    // 32-bit unsigned offsets off uniform base -> SADDR + voffset + imm.
    const unsigned baseIdx = ((p0 + (hiH ? 8u : 0u)) << 4) + (unsigned)col;
    if (p0 + 16u <= (unsigned)n) {          // wave-uniform fast path (scalar branch)
#pragma unroll
      for (int r = 0; r < 8; ++r)
        __builtin_nontemporal_store(acc[r], out + (baseIdx + 16u * r));
    } else {
      const unsigned rowBase = p0 + (hiH ? 8u : 0u);
#pragma unroll
      for (int r = 0; r < 8; ++r)
        if (rowBase + (unsigned)r < (unsigned)n)
          __builtin_nontemporal_store(acc[r], out + (baseIdx + 16u * r));
    }
  }
}

extern "C" void kernel_launch(void* const* d_in, const int* in_sizes, int n_in,
                              void* d_out, int out_size, void* d_ws, size_t ws_size,
                              hipStream_t stream) {
  const float* x = (const float*)d_in[0];
  float* out = (float*)d_out;
  const int n = in_sizes[0] / 3;   // 4,000,000 points
  const int threads = 256;         // 8 waves/block
  const int blocks = 2048;         // 16384 waves, grid-stride over 250k tiles
  SphericalHarmonics_57440892617149_kernel<<<blocks, threads, 0, stream>>>(x, out, n);
}